// RGCNmodel_45595372814388
// MI455X (gfx1250) — compile-verified
//
#include <hip/hip_runtime.h>
#include <hip/hip_bf16.h>

#define N_NODES   50000
#define N_EDGES   800000
#define C_IN      128
#define H_DIM     192
#define N_REL     3
#define N_GRAPHS  64
#define GN_EPS    1e-5f

// GEMM tiling
#define KCHUNK    32
#define APITCH    36                 // 32 + 4 pad: conflict-free ds_load_b64 frags
#define BPITCH    36
#define RT_PER_BLK 5                 // 3125 row tiles = 625 blocks * 5 waves
#define GEMM_THREADS (RT_PER_BLK * 32)
#define TILES_N   (H_DIM / 16)       // 12

typedef __attribute__((ext_vector_type(2))) float v2f;
typedef __attribute__((ext_vector_type(8))) float v8f;

__device__ __forceinline__ void atomAddF(float* p, float v) {
  unsafeAtomicAdd(p, v);   // global_atomic_add_f32
}

// ---------------------------------------------------------------------------
// Stacked weight build:  wcat[k, o],  k in [0, 4*cin)
//   k <  3*cin : W[r,i,o] = sum_b comp[r,b] * bases[b,i,o]
//   k >= 3*cin : root[i,o]
// ---------------------------------------------------------------------------
__global__ void build_wcat(const float* __restrict__ comp,
                           const float* __restrict__ bases,
                           const float* __restrict__ root,
                           float* __restrict__ wcat, int cin) {
  int idx = blockIdx.x * blockDim.x + threadIdx.x;
  int K = 4 * cin;
  if (idx >= K * H_DIM) return;
  int k = idx / H_DIM, o = idx % H_DIM;
  float v;
  if (k < 3 * cin) {
    int r = k / cin, i = k % cin;
    v = 0.f;
#pragma unroll
    for (int b = 0; b < 3; ++b)
      v += comp[r * 3 + b] * bases[((size_t)b * cin + i) * H_DIM + o];
  } else {
    v = root[(size_t)(k - 3 * cin) * H_DIM + o];
  }
  wcat[(size_t)k * H_DIM + o] = v;
}

// --------------------------- edge-degree counts ----------------------------
__global__ void edge_count(const int* __restrict__ dst, const int* __restrict__ et,
                           float* __restrict__ cnt) {
  int e = blockIdx.x * blockDim.x + threadIdx.x;
  if (e >= N_EDGES) return;
  atomAddF(&cnt[dst[e] * N_REL + et[e]], 1.0f);
}

// ---------------- edge gather + scatter-add (wave per edge) ----------------
// xcat row pitch = 4*cin, relation slot t occupies columns [t*cin, (t+1)*cin)
__global__ void __launch_bounds__(256) edge_scatter(
    const float* __restrict__ h, const int* __restrict__ src,
    const int* __restrict__ dst, const int* __restrict__ et,
    float* __restrict__ xcat, int cin, int numWaves) {
  int gw = (int)((blockIdx.x * blockDim.x + threadIdx.x) >> 5);
  int lane = threadIdx.x & 31;
  for (int e = gw; e < N_EDGES; e += numWaves) {
    int s = src[e], d = dst[e], t = et[e];
    const float* hp = h + (size_t)s * cin;
    float* op = xcat + (size_t)d * (4 * cin) + t * cin;
    for (int c = lane; c < cin; c += 32)
      atomAddF(&op[c], hp[c]);
  }
}

// -------- per-relation mean + append h as the "root" K-slice ---------------
__global__ void finish_xcat(const float* __restrict__ h, const float* __restrict__ cnt,
                            float* __restrict__ xcat, int cin) {
  int idx = blockIdx.x * blockDim.x + threadIdx.x;
  int K = 4 * cin;
  if (idx >= N_NODES * K) return;      // <= 38.4M, fits int32
  int n = idx / K, k = idx % K;
  if (k < 3 * cin) {
    float c = cnt[n * N_REL + (k / cin)];
    xcat[idx] = xcat[idx] / fmaxf(c, 1.0f);
  } else {
    xcat[idx] = h[(size_t)n * cin + (k - 3 * cin)];
  }
}

// ---------------------------------------------------------------------------
// Node-update GEMM via V_WMMA_F32_16X16X4_F32, LDS-staged operands.
//   out[N, 192] = xcat[N, K] @ wcat[K, 192] + bias
// Block = 5 waves; wave w owns row tile (blockIdx.x*5 + w) and accumulates all
// 12 column tiles (12 x v8f). Per 32-wide K chunk the block cooperatively
// stages A (80 x 32, pitch 36) and B transposed (192 cols x 32 k, pitch 36)
// into LDS; fragments are read as conflict-free ds_load_b64.
// A frag (16x4): lane L -> row L&15, K = kk + (L>=16 ? 2 : 0) + {x,y}
// B frag (4x16): lane L -> col L&15, same K split across lane halves
// C/D (16x16): vgpr p -> row p + (L>=16 ? 8 : 0), col L&15
// ---------------------------------------------------------------------------
__global__ void __launch_bounds__(GEMM_THREADS) node_gemm_wmma(
    const float* __restrict__ xcat, const float* __restrict__ wcat,
    const float* __restrict__ bias, float* __restrict__ out, int K) {
  __shared__ float sA[RT_PER_BLK * 16 * APITCH];   // 80 x 36
  __shared__ float sB[H_DIM * BPITCH];             // Bt: [col][k], 192 x 36

  const int wave  = threadIdx.x >> 5;
  const int lane  = threadIdx.x & 31;
  const int l15   = lane & 15;
  const int khalf = (lane >> 4) << 1;              // 0 or 2
  const int rowBase = blockIdx.x * (RT_PER_BLK * 16);

  v8f zero = {};
  v8f acc[TILES_N];
#pragma unroll
  for (int t = 0; t < TILES_N; ++t) acc[t] = zero;

  for (int kc = 0; kc < K; kc += KCHUNK) {
    __syncthreads();
    // ---- stage A chunk: 80 rows x 32 k, float4 global loads ----
    {
      int q = threadIdx.x;                         // 640 quads total
#pragma unroll
      for (int it = 0; it < (RT_PER_BLK * 16 * KCHUNK / 4) / GEMM_THREADS;
           ++it, q += GEMM_THREADS) {
        int r  = q >> 3;                           // 8 quads per row
        int k4 = (q & 7) << 2;
        const float4 v = *(const float4*)(xcat + (size_t)(rowBase + r) * K + kc + k4);
        float* dp = &sA[r * APITCH + k4];
        dp[0] = v.x; dp[1] = v.y; dp[2] = v.z; dp[3] = v.w;
      }
    }
    // ---- stage B chunk transposed: Bt[col][kk] ----
    for (int idx = threadIdx.x; idx < KCHUNK * H_DIM; idx += GEMM_THREADS) {
      int kk  = idx / H_DIM;
      int col = idx % H_DIM;
      sB[col * BPITCH + kk] = wcat[(size_t)(kc + kk) * H_DIM + col];
    }
    __syncthreads();
    // ---- compute: 8 k-steps x 12 column tiles ----
    const float* aRow = &sA[(wave * 16 + l15) * APITCH];
#pragma unroll
    for (int kk = 0; kk < KCHUNK; kk += 4) {
      v2f a = *(const v2f*)(aRow + kk + khalf);
#pragma unroll
      for (int t = 0; t < TILES_N; ++t) {
        v2f b = *(const v2f*)(&sB[(t * 16 + l15) * BPITCH + kk + khalf]);
        acc[t] = __builtin_amdgcn_wmma_f32_16x16x4_f32(
            /*neg_a=*/false, a, /*neg_b=*/false, b,
            /*c_mod=*/(short)0, acc[t], /*reuse_a=*/false, /*reuse_b=*/false);
      }
    }
  }
  // ---- epilogue ----
  int rbase = rowBase + wave * 16 + ((lane >> 4) << 3);
#pragma unroll
  for (int t = 0; t < TILES_N; ++t) {
    int col = t * 16 + l15;
    float bo = bias[col];
#pragma unroll
    for (int p = 0; p < 8; ++p)
      out[(size_t)(rbase + p) * H_DIM + col] = acc[t][p] + bo;
  }
}

// --------------------------- graphnorm pieces ------------------------------
__global__ void node_graph_count(const int* __restrict__ batch, float* __restrict__ cntG) {
  int n = blockIdx.x * blockDim.x + threadIdx.x;
  if (n >= N_NODES) return;
  atomAddF(&cntG[batch[n]], 1.0f);
}

__global__ void gn_sum(const float* __restrict__ h, const int* __restrict__ batch,
                       float* __restrict__ gsum) {
  int n = blockIdx.x, c = threadIdx.x;                  // blockDim = H_DIM
  atomAddF(&gsum[batch[n] * H_DIM + c], h[(size_t)n * H_DIM + c]);
}

__global__ void gn_div(float* __restrict__ gsum, const float* __restrict__ cntG) {
  int idx = blockIdx.x * blockDim.x + threadIdx.x;
  if (idx >= N_GRAPHS * H_DIM) return;
  gsum[idx] /= cntG[idx / H_DIM];
}

__global__ void gn_center(float* __restrict__ h, const int* __restrict__ batch,
                          const float* __restrict__ gmean, const float* __restrict__ ms,
                          float* __restrict__ gvar) {
  int n = blockIdx.x, c = threadIdx.x;
  int g = batch[n];
  float v = h[(size_t)n * H_DIM + c] - gmean[g * H_DIM + c] * ms[c];
  h[(size_t)n * H_DIM + c] = v;
  atomAddF(&gvar[g * H_DIM + c], v * v);
}

__global__ void gn_std(float* __restrict__ gvar, const float* __restrict__ cntG) {
  int idx = blockIdx.x * blockDim.x + threadIdx.x;
  if (idx >= N_GRAPHS * H_DIM) return;
  gvar[idx] = sqrtf(gvar[idx] / cntG[idx / H_DIM] + GN_EPS);
}

__global__ void gn_apply_relu(float* __restrict__ h, const int* __restrict__ batch,
                              const float* __restrict__ gstd, const float* __restrict__ w,
                              const float* __restrict__ b) {
  int n = blockIdx.x, c = threadIdx.x;
  int g = batch[n];
  float v = w[c] * h[(size_t)n * H_DIM + c] / gstd[g * H_DIM + c] + b[c];
  h[(size_t)n * H_DIM + c] = fmaxf(v, 0.f);
}

// ------------------------------ pooling / MLP ------------------------------
__global__ void pool_sum(const float* __restrict__ h, const int* __restrict__ batch,
                         float* __restrict__ pool) {
  int n = blockIdx.x, c = threadIdx.x;
  atomAddF(&pool[batch[n] * H_DIM + c], h[(size_t)n * H_DIM + c]);
}

__global__ void pool_div(float* __restrict__ pool, const float* __restrict__ cntG) {
  int idx = blockIdx.x * blockDim.x + threadIdx.x;
  if (idx >= N_GRAPHS * H_DIM) return;
  pool[idx] /= cntG[idx / H_DIM];
}

__global__ void mlp1(const float* __restrict__ g, const float* __restrict__ w1,
                     const float* __restrict__ b1, float* __restrict__ hid) {
  int idx = blockIdx.x * blockDim.x + threadIdx.x;      // 64*96
  if (idx >= N_GRAPHS * (H_DIM / 2)) return;
  int r = idx / (H_DIM / 2), c = idx % (H_DIM / 2);
  float s = b1[c];
  for (int i = 0; i < H_DIM; ++i) s += g[r * H_DIM + i] * w1[i * (H_DIM / 2) + c];
  hid[idx] = fmaxf(s, 0.f);
}

__global__ void mlp2(const float* __restrict__ hid, const float* __restrict__ w2,
                     const float* __restrict__ b2, float* __restrict__ out) {
  int idx = blockIdx.x * blockDim.x + threadIdx.x;      // 64*4
  if (idx >= N_GRAPHS * 4) return;
  int r = idx / 4, c = idx % 4;
  float s = b2[c];
  for (int i = 0; i < H_DIM / 2; ++i) s += hid[r * (H_DIM / 2) + i] * w2[i * 4 + c];
  out[idx] = s;
}

// ===========================================================================
extern "C" void kernel_launch(void* const* d_in, const int* in_sizes, int n_in,
                              void* d_out, int out_size, void* d_ws, size_t ws_size,
                              hipStream_t stream) {
  const float* x      = (const float*)d_in[0];
  const int*   eidx   = (const int*)d_in[1];
  const int*   src    = eidx;
  const int*   dst    = eidx + N_EDGES;
  const int*   etype  = (const int*)d_in[2];
  const int*   batch  = (const int*)d_in[3];
  const float* bases1 = (const float*)d_in[4];
  const float* comp1  = (const float*)d_in[5];
  const float* root1  = (const float*)d_in[6];
  const float* bias1  = (const float*)d_in[7];
  const float* gn1_w  = (const float*)d_in[8];
  const float* gn1_b  = (const float*)d_in[9];
  const float* gn1_ms = (const float*)d_in[10];
  const float* bases2 = (const float*)d_in[11];
  const float* comp2  = (const float*)d_in[12];
  const float* root2  = (const float*)d_in[13];
  const float* bias2  = (const float*)d_in[14];
  const float* gn2_w  = (const float*)d_in[15];
  const float* gn2_b  = (const float*)d_in[16];
  const float* gn2_ms = (const float*)d_in[17];
  const float* w1     = (const float*)d_in[18];
  const float* b1     = (const float*)d_in[19];
  const float* w2     = (const float*)d_in[20];
  const float* b2     = (const float*)d_in[21];

  // ---- workspace partition (256B aligned) ----
  char* wsp = (char*)d_ws;
  size_t off = 0;
  auto salloc = [&](size_t nbytes) -> float* {
    float* p = (float*)(wsp + off);
    off += (nbytes + 255) & ~(size_t)255;
    return p;
  };
  float* xcat  = salloc(sizeof(float) * (size_t)N_NODES * 4 * H_DIM); // reused both layers
  float* h1    = salloc(sizeof(float) * (size_t)N_NODES * H_DIM);
  float* h2    = salloc(sizeof(float) * (size_t)N_NODES * H_DIM);
  float* wcat  = salloc(sizeof(float) * 4 * H_DIM * H_DIM);
  float* cnt   = salloc(sizeof(float) * (size_t)N_NODES * N_REL);
  float* cntG  = salloc(sizeof(float) * N_GRAPHS);
  float* gmean = salloc(sizeof(float) * N_GRAPHS * H_DIM);
  float* gvar  = salloc(sizeof(float) * N_GRAPHS * H_DIM);
  float* pool  = salloc(sizeof(float) * N_GRAPHS * H_DIM);
  float* hid   = salloc(sizeof(float) * N_GRAPHS * (H_DIM / 2));
  (void)ws_size; (void)n_in; (void)in_sizes; (void)out_size;

  const int gemmBlocks = (N_NODES / 16) / RT_PER_BLK;   // 625, exact
  const int GH = N_GRAPHS * H_DIM;
  const int scatterBlocks = 2048;                        // 2048*8 = 16384 waves
  const int scatterWaves = scatterBlocks * (256 / 32);

  // ---- shared precomputation ----
  hipMemsetAsync(cnt, 0, sizeof(float) * (size_t)N_NODES * N_REL, stream);
  hipMemsetAsync(cntG, 0, sizeof(float) * N_GRAPHS, stream);
  edge_count<<<(N_EDGES + 255) / 256, 256, 0, stream>>>(dst, etype, cnt);
  node_graph_count<<<(N_NODES + 255) / 256, 256, 0, stream>>>(batch, cntG);

  // =================== Layer 1: C_IN=128 -> H_DIM=192 =====================
  {
    const int cin = C_IN, K = 4 * C_IN;                 // 512
    build_wcat<<<(K * H_DIM + 255) / 256, 256, 0, stream>>>(comp1, bases1, root1, wcat, cin);
    hipMemsetAsync(xcat, 0, sizeof(float) * (size_t)N_NODES * K, stream);
    edge_scatter<<<scatterBlocks, 256, 0, stream>>>(x, src, dst, etype, xcat, cin, scatterWaves);
    finish_xcat<<<(N_NODES * K + 255) / 256, 256, 0, stream>>>(x, cnt, xcat, cin);
    node_gemm_wmma<<<gemmBlocks, GEMM_THREADS, 0, stream>>>(xcat, wcat, bias1, h1, K);

    hipMemsetAsync(gmean, 0, sizeof(float) * GH, stream);
    hipMemsetAsync(gvar, 0, sizeof(float) * GH, stream);
    gn_sum<<<N_NODES, H_DIM, 0, stream>>>(h1, batch, gmean);
    gn_div<<<(GH + 255) / 256, 256, 0, stream>>>(gmean, cntG);
    gn_center<<<N_NODES, H_DIM, 0, stream>>>(h1, batch, gmean, gn1_ms, gvar);
    gn_std<<<(GH + 255) / 256, 256, 0, stream>>>(gvar, cntG);
    gn_apply_relu<<<N_NODES, H_DIM, 0, stream>>>(h1, batch, gvar, gn1_w, gn1_b);
  }

  // =================== Layer 2: H_DIM=192 -> H_DIM=192 ====================
  {
    const int cin = H_DIM, K = 4 * H_DIM;               // 768
    build_wcat<<<(K * H_DIM + 255) / 256, 256, 0, stream>>>(comp2, bases2, root2, wcat, cin);
    hipMemsetAsync(xcat, 0, sizeof(float) * (size_t)N_NODES * K, stream);
    edge_scatter<<<scatterBlocks, 256, 0, stream>>>(h1, src, dst, etype, xcat, cin, scatterWaves);
    finish_xcat<<<(N_NODES * K + 255) / 256, 256, 0, stream>>>(h1, cnt, xcat, cin);
    node_gemm_wmma<<<gemmBlocks, GEMM_THREADS, 0, stream>>>(xcat, wcat, bias2, h2, K);

    hipMemsetAsync(gmean, 0, sizeof(float) * GH, stream);
    hipMemsetAsync(gvar, 0, sizeof(float) * GH, stream);
    gn_sum<<<N_NODES, H_DIM, 0, stream>>>(h2, batch, gmean);
    gn_div<<<(GH + 255) / 256, 256, 0, stream>>>(gmean, cntG);
    gn_center<<<N_NODES, H_DIM, 0, stream>>>(h2, batch, gmean, gn2_ms, gvar);
    gn_std<<<(GH + 255) / 256, 256, 0, stream>>>(gvar, cntG);
    gn_apply_relu<<<N_NODES, H_DIM, 0, stream>>>(h2, batch, gvar, gn2_w, gn2_b);
  }

  // ====================== pool + MLP head =================================
  hipMemsetAsync(pool, 0, sizeof(float) * GH, stream);
  pool_sum<<<N_NODES, H_DIM, 0, stream>>>(h2, batch, pool);
  pool_div<<<(GH + 255) / 256, 256, 0, stream>>>(pool, cntG);
  mlp1<<<(N_GRAPHS * (H_DIM / 2) + 255) / 256, 256, 0, stream>>>(pool, w1, b1, hid);
  mlp2<<<(N_GRAPHS * 4 + 255) / 256, 256, 0, stream>>>(hid, w2, b2, (float*)d_out);
}